// ContinualLearningModule_71854802862768
// MI455X (gfx1250) — compile-verified
//
#include <hip/hip_runtime.h>
#include <math.h>

typedef __attribute__((ext_vector_type(2))) float v2f;
typedef __attribute__((ext_vector_type(8))) float v8f;

#define TOPK 5
#define EPS 1e-8f
#define TKA_BLOCKS 128
#define TKA_THREADS 256

// -------- helper: pick c[r] for r in 0..7 via select chain (no scratch) ----
__device__ __forceinline__ float pick8(v8f c, int r) {
    float v = c[0];
    v = (r == 1) ? c[1] : v;
    v = (r == 2) ? c[2] : v;
    v = (r == 3) ? c[3] : v;
    v = (r == 4) ? c[4] : v;
    v = (r == 5) ? c[5] : v;
    v = (r == 6) ? c[6] : v;
    v = (r == 7) ? c[7] : v;
    return v;
}

// ---------------------------------------------------------------------------
// Kernel 1: importance MLP + query norm^2.  One block of H (=128) threads.
// Writes: w_scal[0]=importance, [1]=||cur||^2, d_out[0]=importance,
//         sims[N] = stored ? 1.0 (self cosine) : -inf  (masked candidate).
// ---------------------------------------------------------------------------
__global__ void imp_kernel(const float* __restrict__ cur, const float* __restrict__ tgt,
                           const float* __restrict__ W1, const float* __restrict__ b1,
                           const float* __restrict__ W2, const float* __restrict__ b2,
                           int D, int H, int N,
                           float* __restrict__ w_scal, float* __restrict__ w_sims,
                           float* __restrict__ out) {
    __shared__ float sh[256];
    int j = threadIdx.x;            // 0..H-1
    // h_j = relu(b1[j] + sum_i in[i] * W1[i*H+j]) ; in = concat(cur, tgt)
    float acc = b1[j];
    for (int i = 0; i < D; ++i)       acc += cur[i] * W1[(size_t)i * H + j];
    for (int i = 0; i < D; ++i)       acc += tgt[i] * W1[(size_t)(D + i) * H + j];
    acc = fmaxf(acc, 0.0f);
    sh[j] = acc * W2[j];
    __syncthreads();
    for (int s = H >> 1; s > 0; s >>= 1) {
        if (j < s) sh[j] += sh[j + s];
        __syncthreads();
    }
    float imp = 1.0f / (1.0f + expf(-(sh[0] + b2[0])));
    __syncthreads();
    // ||cur||^2
    float q = 0.0f;
    for (int i = j; i < D; i += H) { float c = cur[i]; q += c * c; }
    sh[j] = q;
    __syncthreads();
    for (int s = H >> 1; s > 0; s >>= 1) {
        if (j < s) sh[j] += sh[j + s];
        __syncthreads();
    }
    if (j == 0) {
        float qsq = sh[0];
        w_scal[0] = imp;
        w_scal[1] = qsq;
        // self-candidate: cos(cur,cur) = qsq / max(qn*qn, EPS) (==1 for our data)
        float qn = sqrtf(qsq);
        float self_sim = qsq / fmaxf(qn * qn, EPS);
        w_sims[N] = (imp > 0.5f) ? self_sim : -INFINITY;
        out[0] = imp;
    }
}

// ---------------------------------------------------------------------------
// Kernel 2: WMMA f32 16x16x4 similarity scan.  Each wave owns 16 rows.
// A tile (16x4): lane l -> row = l&15; lane half picks K offset 0 or 2
//   (ISA 7.12.2: lanes 0-15 VGPR0/1 = K0/K1, lanes 16-31 = K2/K3).
// B (4x16): query chunk broadcast to all 16 columns -> every column of D
//   equals dot(row, cur).  Row m's dot extracted via pick8 + shfl.
// A rows streamed with TH=NT (read-once, 400MB > 192MB L2: don't thrash L2,
// keep the sims buffer hot for the top-k passes).
// ---------------------------------------------------------------------------
__global__ void sim_kernel(const float* __restrict__ mem, const float* __restrict__ cur,
                           const float* __restrict__ w_scal,
                           int D, int ntiles,
                           float* __restrict__ w_sims) {
    int wave = threadIdx.x >> 5;
    int lane = threadIdx.x & 31;
    int tile = blockIdx.x * (blockDim.x >> 5) + wave;
    if (tile >= ntiles) return;                      // wave-uniform: EXEC all-1 inside
    float qn = sqrtf(w_scal[1]);
    int row0 = tile << 4;
    int m    = lane & 15;
    int koff = (lane >> 4) << 1;                     // 0 (lanes 0-15) or 2 (16-31)
    const float* arow = mem + (size_t)(row0 + m) * D;
    v8f c = {};
    float sq = 0.0f;
    for (int k0 = 0; k0 < D; k0 += 4) {
        v2f a = __builtin_nontemporal_load((const v2f*)(arow + k0 + koff));
        v2f b;
        b.x = cur[k0 + koff];
        b.y = cur[k0 + koff + 1];
        sq += a.x * a.x + a.y * a.y;
        c = __builtin_amdgcn_wmma_f32_16x16x4_f32(false, a, false, b,
                                                  (short)0, c, false, false);
    }
    sq += __shfl_xor(sq, 16, 32);                    // full row sumsq in lanes 0-15
    float sel = pick8(c, lane & 7);                  // low lanes: rows 0-7; high: 8-15
    int   src = (lane & 8) ? (lane + 8) : lane;      // lane L<16 fetches row L's dot
    float dot = __shfl(sel, src, 32);
    if (lane < 16) {
        float denom = fmaxf(sqrtf(sq) * qn, EPS);
        w_sims[row0 + lane] = dot / denom;
    }
}

// Scalar tail for N % 16 rows (N=100000 is divisible by 16; kept for safety).
__global__ void sim_tail_kernel(const float* __restrict__ mem, const float* __restrict__ cur,
                                const float* __restrict__ w_scal,
                                int D, int row_start, int N,
                                float* __restrict__ w_sims) {
    int r = row_start + threadIdx.x;
    if (r >= N) return;
    float qn = sqrtf(w_scal[1]);
    const float* row = mem + (size_t)r * D;
    float d = 0.0f, s = 0.0f;
    for (int k = 0; k < D; ++k) { float a = row[k]; d += a * cur[k]; s += a * a; }
    w_sims[r] = d / fmaxf(sqrtf(s) * qn, EPS);
}

// ---------------------------------------------------------------------------
// Kernel 3a: hierarchical top-5, stage A.  Each block finds the exact top-5
// of its contiguous chunk (global top-5 is within the union of local top-5s).
// Chunk (~3KB) stays in L0 across the 5 sweeps.
// ---------------------------------------------------------------------------
__global__ void topk_stage_a(const float* __restrict__ sims, int Ncand,
                             float* __restrict__ cand_val, int* __restrict__ cand_idx) {
    __shared__ float sval[TKA_THREADS];
    __shared__ int   sidx[TKA_THREADS];
    __shared__ int   chosen[TOPK];
    int t = threadIdx.x;
    int chunk = (Ncand + gridDim.x - 1) / gridDim.x;
    int lo = blockIdx.x * chunk;
    int hi = min(lo + chunk, Ncand);
    for (int r = 0; r < TOPK; ++r) {
        float bestv = -INFINITY;
        int   besti = -1;
        for (int i = lo + t; i < hi; i += blockDim.x) {
            bool skip = false;
            for (int p = 0; p < r; ++p) skip = skip || (chosen[p] == i);
            if (skip) continue;
            float s = sims[i];
            if (s > bestv) { bestv = s; besti = i; }
        }
        sval[t] = bestv;
        sidx[t] = besti;
        __syncthreads();
        for (int s2 = blockDim.x >> 1; s2 > 0; s2 >>= 1) {
            if (t < s2 && sval[t + s2] > sval[t]) { sval[t] = sval[t + s2]; sidx[t] = sidx[t + s2]; }
            __syncthreads();
        }
        if (t == 0) {
            chosen[r] = sidx[0];
            cand_val[blockIdx.x * TOPK + r] = sval[0];
            cand_idx[blockIdx.x * TOPK + r] = sidx[0];
        }
        __syncthreads();
    }
}

// Kernel 3b: stage B — exact top-5 over the 640 stage-A candidates.
__global__ void topk_stage_b(const float* __restrict__ cand_val,
                             const int* __restrict__ cand_idx, int M,
                             int* __restrict__ top_idx, float* __restrict__ top_sim) {
    __shared__ float sval[256];
    __shared__ int   sidx[256];      // candidate-slot, not memory index
    __shared__ int   chosen[TOPK];
    int t = threadIdx.x;
    for (int r = 0; r < TOPK; ++r) {
        float bestv = -INFINITY;
        int   bests = -1;
        for (int i = t; i < M; i += blockDim.x) {
            bool skip = false;
            for (int p = 0; p < r; ++p) skip = skip || (chosen[p] == i);
            if (skip) continue;
            float s = cand_val[i];
            if (s > bestv) { bestv = s; bests = i; }
        }
        sval[t] = bestv;
        sidx[t] = bests;
        __syncthreads();
        for (int s2 = blockDim.x >> 1; s2 > 0; s2 >>= 1) {
            if (t < s2 && sval[t + s2] > sval[t]) { sval[t] = sval[t + s2]; sidx[t] = sidx[t + s2]; }
            __syncthreads();
        }
        if (t == 0) {
            chosen[r]  = sidx[0];
            top_idx[r] = cand_idx[sidx[0]];
            top_sim[r] = sval[0];
        }
        __syncthreads();
    }
}

// ---------------------------------------------------------------------------
// Kernel 4: gather top-5 rows, mean -> w_m, reg loss -> out[1+D].
// One block of D (=1024) threads; per-k distances via shared ds_add_f32.
// ---------------------------------------------------------------------------
__global__ void gather_kernel(const float* __restrict__ cur, const float* __restrict__ mem,
                              const float* __restrict__ mem_imp,
                              const float* __restrict__ w_scal,
                              const int* __restrict__ top_idx,
                              const float* __restrict__ reg_ctrl,
                              int D, int N,
                              float* __restrict__ w_m, float* __restrict__ out) {
    __shared__ float sdist[TOPK];
    __shared__ float simp[TOPK];
    int t = threadIdx.x;
    if (t < TOPK) {
        sdist[t] = 0.0f;
        int idx = top_idx[t];
        simp[t] = (idx < N) ? mem_imp[idx] : w_scal[0];
    }
    __syncthreads();
    float c = cur[t];
    float s = 0.0f;
    for (int k = 0; k < TOPK; ++k) {
        int idx = top_idx[k];
        float f = (idx < N) ? mem[(size_t)idx * D + t] : c;   // self-candidate = cur
        s += f;
        float d = c - f;
        atomicAdd(&sdist[k], d * d);
    }
    w_m[t] = s * (1.0f / TOPK);
    __syncthreads();
    if (t == 0) {
        float rl = 0.0f;
        for (int k = 0; k < TOPK; ++k) rl += simp[k] * sdist[k];
        out[1 + D] = reg_ctrl[0] * rl * (1.0f / TOPK);
    }
}

// ---------------------------------------------------------------------------
// Kernel 5/6: WMMA GEMV  out[j] = act(bias[j] + sum_k x[k] * W[k*n_out + j]).
// A tile: A[m][k] = W[(k0+k)*n_out + j0+m]  (m = output within 16-tile).
// ---------------------------------------------------------------------------
__global__ void gemv_wmma_kernel(const float* __restrict__ W, const float* __restrict__ x,
                                 const float* __restrict__ bias, int n_in, int n_out,
                                 int do_relu, float* __restrict__ out) {
    int wave = threadIdx.x >> 5;
    int lane = threadIdx.x & 31;
    int tile = blockIdx.x * (blockDim.x >> 5) + wave;
    if (tile * 16 >= n_out) return;                  // wave-uniform guard
    int j0   = tile << 4;
    int m    = lane & 15;
    int koff = (lane >> 4) << 1;
    v8f c = {};
    for (int k0 = 0; k0 < n_in; k0 += 4) {
        v2f a, b;
        a.x = W[(size_t)(k0 + koff)     * n_out + j0 + m];
        a.y = W[(size_t)(k0 + koff + 1) * n_out + j0 + m];
        b.x = x[k0 + koff];
        b.y = x[k0 + koff + 1];
        c = __builtin_amdgcn_wmma_f32_16x16x4_f32(false, a, false, b,
                                                  (short)0, c, false, false);
    }
    float sel = pick8(c, lane & 7);
    int   src = (lane & 8) ? (lane + 8) : lane;
    float v = __shfl(sel, src, 32);
    if (lane < 16) {
        v += bias[j0 + lane];
        if (do_relu) v = fmaxf(v, 0.0f);
        out[j0 + lane] = v;
    }
}

// ---------------------------------------------------------------------------
extern "C" void kernel_launch(void* const* d_in, const int* in_sizes, int n_in_bufs,
                              void* d_out, int out_size, void* d_ws, size_t ws_size,
                              hipStream_t stream) {
    const float* cur     = (const float*)d_in[0];
    const float* tgt     = (const float*)d_in[1];
    const float* mem     = (const float*)d_in[2];
    const float* mem_imp = (const float*)d_in[3];
    const float* W1      = (const float*)d_in[4];
    const float* b1      = (const float*)d_in[5];
    const float* W2      = (const float*)d_in[6];
    const float* b2      = (const float*)d_in[7];
    const float* Wc1     = (const float*)d_in[8];
    const float* bc1     = (const float*)d_in[9];
    const float* Wc2     = (const float*)d_in[10];
    const float* bc2     = (const float*)d_in[11];
    const float* reg_c   = (const float*)d_in[12];

    const int D  = in_sizes[0];     // 1024
    const int N  = in_sizes[3];     // 100000
    const int H1 = in_sizes[5];     // 128
    const int C1 = in_sizes[9];     // 256
    float* out = (float*)d_out;     // [0]=importance, [1..D]=consolidated, [1+D]=reg

    // workspace layout (floats)
    float* w      = (float*)d_ws;
    float* w_scal = w;                               // 16 scalars
    int    Np     = ((N + 1 + 7) / 8) * 8;
    float* w_sims = w + 16;                          // N+1 final cosine sims
    float* w_cval = w_sims + Np;                     // TKA_BLOCKS*TOPK
    int*   w_cidx = (int*)(w_cval + TKA_BLOCKS * TOPK);
    int*   w_tidx = (int*)(w_cidx + TKA_BLOCKS * TOPK);   // 8
    float* w_tsim = (float*)(w_tidx + 8);            // 8
    float* w_m    = w_tsim + 8;                      // D
    float* w_h2   = w_m + D;                         // C1

    // 1) importance MLP + ||cur||^2 + self-candidate sim slot
    imp_kernel<<<1, H1, 0, stream>>>(cur, tgt, W1, b1, W2, b2,
                                     D, H1, N, w_scal, w_sims, out);

    // 2) similarity scan (bandwidth-bound 400MB NT stream through WMMA f32)
    int ntiles = N >> 4;
    if (ntiles > 0)
        sim_kernel<<<(ntiles + 7) / 8, 256, 0, stream>>>(mem, cur, w_scal,
                                                         D, ntiles, w_sims);
    int rem_start = ntiles << 4;
    if (rem_start < N)
        sim_tail_kernel<<<1, 16, 0, stream>>>(mem, cur, w_scal,
                                              D, rem_start, N, w_sims);

    // 3) hierarchical exact top-5
    topk_stage_a<<<TKA_BLOCKS, TKA_THREADS, 0, stream>>>(w_sims, N + 1,
                                                         w_cval, w_cidx);
    topk_stage_b<<<1, 256, 0, stream>>>(w_cval, w_cidx, TKA_BLOCKS * TOPK,
                                        w_tidx, w_tsim);

    // 4) gather + mean + reg loss
    gather_kernel<<<1, D, 0, stream>>>(cur, mem, mem_imp, w_scal, w_tidx,
                                       reg_c, D, N, w_m, out);

    // 5) consolidation layer 1: [D] @ [D,C1] -> relu -> w_h2
    {
        int tiles = (C1 + 15) / 16;
        int blocks = (tiles + 15) / 16;
        gemv_wmma_kernel<<<blocks, 512, 0, stream>>>(Wc1, w_m, bc1, D, C1, 1, w_h2);
    }
    // 6) consolidation layer 2: [C1] @ [C1,D] -> out[1..D]
    {
        int tiles = (D + 15) / 16;
        int blocks = (tiles + 15) / 16;
        gemv_wmma_kernel<<<blocks, 512, 0, stream>>>(Wc2, w_h2, bc2, C1, D, 0, out + 1);
    }
    (void)n_in_bufs; (void)out_size; (void)ws_size;
}